// Model_67551245632178
// MI455X (gfx1250) — compile-verified
//
#include <hip/hip_runtime.h>
#include <math.h>

typedef float v2f __attribute__((ext_vector_type(2)));
typedef float v8f __attribute__((ext_vector_type(8)));

#define CH 32
#define NGRAPH 128
#define TB 256

static inline int cdiv(int a, int b) { return (a + b - 1) / b; }

// ---------------- degree / normalization ----------------
__global__ void k_deg_init(float* __restrict__ deg, int n) {
    int i = blockIdx.x * blockDim.x + threadIdx.x;
    if (i < n) deg[i] = 1.0f;  // self-loop contributes 1
}

__global__ void k_deg_accum(const int* __restrict__ dst, float* __restrict__ deg, int e) {
    int i = blockIdx.x * blockDim.x + threadIdx.x;
    if (i < e) atomicAdd(&deg[dst[i]], 1.0f);
}

__global__ void k_dinv(float* __restrict__ deg, int n) {
    int i = blockIdx.x * blockDim.x + threadIdx.x;
    if (i < n) deg[i] = rsqrtf(deg[i]);  // in place: deg -> deg^{-1/2}
}

// ---------------- embedding ----------------
__global__ void k_embed(const int* __restrict__ x_idx, const float* __restrict__ embed_w,
                        float* __restrict__ h, int n) {
    int t = blockIdx.x * blockDim.x + threadIdx.x;
    if (t < n * CH) {
        int node = t >> 5;
        int c    = t & 31;
        h[t] = embed_w[x_idx[node] * CH + c];
    }
}

// ---------------- fused GCN transform via V_WMMA_F32_16X16X4_F32 ----------------
// One wave per 16-node tile. Fusions:
//   A-load:  (HAS_BIAS) x = relu(agg_prev + bias_prev)   [previous layer epilogue]
//   D-store: z = x @ W^T  AND  agg = dinv^2 * z          [self-loop init, in place]
// In-place h/agg overwrite is safe: each wave reads then writes only its own rows.
template <bool HAS_BIAS>
__global__ void k_conv_matmul(const float* hin, const float* __restrict__ W,
                              const float* __restrict__ bias_prev,
                              const float* __restrict__ dinv,
                              float* __restrict__ z, float* agg, int n) {
    const int lane   = threadIdx.x & 31;
    const int wave   = threadIdx.x >> 5;
    const int tile   = blockIdx.x * (blockDim.x >> 5) + wave;
    const int ntiles = (n + 15) >> 4;
    if (tile >= ntiles) return;

    const int n0   = tile << 4;
    const int half = lane >> 4;   // 0: K 0/1, 1: K 2/3 (ISA 16x4 f32 A layout)
    const int l15  = lane & 15;
    const int koff = half * 2;

    int arow = n0 + l15;
    if (arow >= n) arow = n - 1;  // clamp reads on a partial last tile

    const float* ap  = hin + (size_t)arow * CH + koff;          // A: h[node][k]
    const float* bp0 = W + (size_t)l15 * CH + koff;             // B jt=0: W[j][k]
    const float* bp1 = W + (size_t)(16 + l15) * CH + koff;      // B jt=1

    v8f acc0 = {};
    v8f acc1 = {};
#pragma unroll
    for (int kk = 0; kk < 8; ++kk) {
        v2f a = *(const v2f*)(ap + kk * 4);
        if (HAS_BIAS) {
            v2f bb = *(const v2f*)(bias_prev + kk * 4 + koff);
            a.x = fmaxf(a.x + bb.x, 0.0f);
            a.y = fmaxf(a.y + bb.y, 0.0f);
        }
        v2f b0 = *(const v2f*)(bp0 + kk * 4);
        v2f b1 = *(const v2f*)(bp1 + kk * 4);
        acc0 = __builtin_amdgcn_wmma_f32_16x16x4_f32(false, a, false, b0,
                                                     (short)0, acc0, false, false);
        acc1 = __builtin_amdgcn_wmma_f32_16x16x4_f32(false, a, false, b1,
                                                     (short)0, acc1, false, false);
    }

    // D layout: VGPR v holds row M=v (lanes 0-15) / M=v+8 (lanes 16-31), N=l15
    if (n0 + 16 <= n) {  // fast path: full tile, no per-row guards
#pragma unroll
        for (int v = 0; v < 8; ++v) {
            int row  = n0 + v + half * 8;
            float di = dinv[row];
            float s  = di * di;
            float* zp = z   + (size_t)row * CH + l15;
            float* gp = agg + (size_t)row * CH + l15;
            zp[0]  = acc0[v];
            zp[16] = acc1[v];
            gp[0]  = s * acc0[v];
            gp[16] = s * acc1[v];
        }
    } else {
#pragma unroll
        for (int v = 0; v < 8; ++v) {
            int row = n0 + v + half * 8;
            if (row < n) {
                float di = dinv[row];
                float s  = di * di;
                float* zp = z   + (size_t)row * CH + l15;
                float* gp = agg + (size_t)row * CH + l15;
                zp[0]  = acc0[v];
                zp[16] = acc1[v];
                gp[0]  = s * acc0[v];
                gp[16] = s * acc1[v];
            }
        }
    }
}

// ---------------- edge scatter: wave per edge, lane per channel ----------------
__global__ void k_scatter(const int* __restrict__ src, const int* __restrict__ dst,
                          const float* __restrict__ dinv, const float* __restrict__ z,
                          float* __restrict__ agg, int e) {
    int gid  = blockIdx.x * blockDim.x + threadIdx.x;
    int edge = gid >> 5;        // wave-uniform (256-thread blocks are wave-aligned)
    int lane = gid & 31;
    if (edge >= e) return;
    int eU = __builtin_amdgcn_readfirstlane(edge);   // force SGPR -> scalar loads
    int s  = src[eU];
    int d  = dst[eU];
    float nrm = dinv[s] * dinv[d];
    float val = nrm * z[(size_t)s * CH + lane];
    atomicAdd(&agg[(size_t)d * CH + lane], val);
}

// ---------------- pooling: final bias+relu fused with segment_max ----------------
__global__ void k_pool_init(float* __restrict__ pooled) {
    int t = blockIdx.x * blockDim.x + threadIdx.x;
    if (t < NGRAPH * CH) pooled[t] = 0.0f;  // empty graphs -> 0 (matches isfinite guard)
}

__global__ void k_bias_relu_pool(const int* __restrict__ batch, const float* __restrict__ bias,
                                 const float* __restrict__ agg, float* __restrict__ pooled,
                                 int n) {
    int gid  = blockIdx.x * blockDim.x + threadIdx.x;
    int node = gid >> 5;        // wave-uniform
    int lane = gid & 31;
    if (node >= n) return;
    int nU = __builtin_amdgcn_readfirstlane(node);
    int g  = batch[nU];
    float v = fmaxf(agg[(size_t)nU * CH + lane] + bias[lane], 0.0f);
    // v >= 0 so unsigned bit-pattern ordering == float ordering
    atomicMax((unsigned int*)&pooled[g * CH + lane], __float_as_uint(v));
}

// ---------------- MLP head + log_softmax: one thread per graph ----------------
__global__ void k_head(const float* __restrict__ pooled,
                       const float* __restrict__ d0_w, const float* __restrict__ d0_b,
                       const float* __restrict__ dense_w, const float* __restrict__ dense_b,
                       const float* __restrict__ fin_w, const float* __restrict__ fin_b,
                       float* __restrict__ out) {
    int g = threadIdx.x;
    if (g >= NGRAPH) return;
    float x[CH], y[CH];
#pragma unroll
    for (int c = 0; c < CH; ++c) x[c] = pooled[g * CH + c];

    // d0
    for (int j = 0; j < CH; ++j) {
        float s = d0_b[j];
        for (int k = 0; k < CH; ++k) s += x[k] * d0_w[j * CH + k];
        y[j] = fmaxf(s, 0.0f);
    }
    for (int c = 0; c < CH; ++c) x[c] = y[c];

    // 3 dense layers
    for (int l = 0; l < 3; ++l) {
        const float* W = dense_w + l * CH * CH;
        const float* B = dense_b + l * CH;
        for (int j = 0; j < CH; ++j) {
            float s = B[j];
            for (int k = 0; k < CH; ++k) s += x[k] * W[j * CH + k];
            y[j] = fmaxf(s, 0.0f);
        }
        for (int c = 0; c < CH; ++c) x[c] = y[c];
    }

    // final 32 -> 2 + log_softmax
    float l0 = fin_b[0], l1 = fin_b[1];
    for (int k = 0; k < CH; ++k) {
        l0 += x[k] * fin_w[k];
        l1 += x[k] * fin_w[CH + k];
    }
    float m   = fmaxf(l0, l1);
    float lse = m + logf(expf(l0 - m) + expf(l1 - m));
    out[g * 2 + 0] = l0 - lse;
    out[g * 2 + 1] = l1 - lse;
}

extern "C" void kernel_launch(void* const* d_in, const int* in_sizes, int n_in,
                              void* d_out, int out_size, void* d_ws, size_t ws_size,
                              hipStream_t stream) {
    const int*   x_idx   = (const int*)d_in[0];
    const int*   eidx    = (const int*)d_in[1];
    const int*   batch   = (const int*)d_in[2];
    const float* embed_w = (const float*)d_in[3];
    const float* conv_w  = (const float*)d_in[4];
    const float* conv_b  = (const float*)d_in[5];
    const float* d0_w    = (const float*)d_in[6];
    const float* d0_b    = (const float*)d_in[7];
    const float* dense_w = (const float*)d_in[8];
    const float* dense_b = (const float*)d_in[9];
    const float* fin_w   = (const float*)d_in[10];
    const float* fin_b   = (const float*)d_in[11];

    const int n = in_sizes[0];
    const int e = in_sizes[1] / 2;
    const int* src = eidx;        // edge_index[0]
    const int* dst = eidx + e;    // edge_index[1]

    // workspace carve-up (256B aligned)
    char*  ws  = (char*)d_ws;
    size_t off = 0;
    auto carve = [&](size_t bytes) -> void* {
        void* p = ws + off;
        off = (off + bytes + 255) & ~(size_t)255;
        return p;
    };
    float* dinv   = (float*)carve((size_t)n * sizeof(float));
    float* h      = (float*)carve((size_t)n * CH * sizeof(float));  // doubles as agg
    float* z      = (float*)carve((size_t)n * CH * sizeof(float));
    float* pooled = (float*)carve((size_t)NGRAPH * CH * sizeof(float));

    const int nc = n * CH;

    // normalization
    k_deg_init<<<cdiv(n, TB), TB, 0, stream>>>(dinv, n);
    k_deg_accum<<<cdiv(e, TB), TB, 0, stream>>>(dst, dinv, e);
    k_dinv<<<cdiv(n, TB), TB, 0, stream>>>(dinv, n);

    // embedding
    k_embed<<<cdiv(nc, TB), TB, 0, stream>>>(x_idx, embed_w, h, n);

    // 5 GCN layers: WMMA transform (+fused prev epilogue, +fused self-loop init), edge scatter
    const int ntiles      = (n + 15) >> 4;
    const int mm_blocks   = cdiv(ntiles, TB / 32);
    const int scat_blocks = cdiv(e * 32, TB);
    for (int l = 0; l < 5; ++l) {
        if (l == 0)
            k_conv_matmul<false><<<mm_blocks, TB, 0, stream>>>(
                h, conv_w, nullptr, dinv, z, h, n);
        else
            k_conv_matmul<true><<<mm_blocks, TB, 0, stream>>>(
                h, conv_w + l * CH * CH, conv_b + (l - 1) * CH, dinv, z, h, n);
        k_scatter<<<scat_blocks, TB, 0, stream>>>(src, dst, dinv, z, h, e);
    }

    // pool (final bias+relu fused) + head
    k_pool_init<<<cdiv(NGRAPH * CH, TB), TB, 0, stream>>>(pooled);
    k_bias_relu_pool<<<cdiv(n * 32, TB), TB, 0, stream>>>(batch, conv_b + 4 * CH, h, pooled, n);
    k_head<<<1, NGRAPH, 0, stream>>>(pooled, d0_w, d0_b, dense_w, dense_b,
                                     fin_w, fin_b, (float*)d_out);
}